// BahadanauAttn_85624468013571
// MI455X (gfx1250) — compile-verified
//
#include <hip/hip_runtime.h>
#include <hip/hip_bf16.h>
#include <math.h>

// Sizes fixed by the reference.
#define HDIM 4096
#define SDIM 8192
#define HSPLIT 16          // phase-1 h-chunks
#define KSPLIT 4           // phase-2 K-chunks per row tile
#define KCHUNK (HDIM / KSPLIT)   // 1024

typedef __attribute__((ext_vector_type(2))) float v2f;
typedef __attribute__((ext_vector_type(8))) float v8f;

// ---------------------------------------------------------------------------
// Phase 1: part[hs][k] = sum_{h in chunk hs} other[h] * W[h, HDIM + k]
// grid = (16 k-chunks, 16 h-chunks) x 256 threads = 2048 waves streaming the
// right half of W (64 MB) fully coalesced; unroll 32 for deep load pipelining.
// ---------------------------------------------------------------------------
__global__ void __launch_bounds__(256)
u2_partial_kernel(const float* __restrict__ W, const float* __restrict__ other,
                  float* __restrict__ part) {
  const int k  = blockIdx.x * 256 + threadIdx.x;   // [0, 4096)
  const int hs = blockIdx.y;                       // [0, HSPLIT)
  const int h0 = hs * (HDIM / HSPLIT);
  float acc = 0.0f;
  const float* __restrict__ wp = W + (size_t)h0 * (2 * HDIM) + HDIM + k;
#pragma unroll 32
  for (int h = 0; h < HDIM / HSPLIT; ++h) {
    acc = fmaf(other[h0 + h], *wp, acc);
    wp += 2 * HDIM;
  }
  part[hs * HDIM + k] = acc;
}

// ---------------------------------------------------------------------------
// Phase 1b: u2[k] = sum over HSPLIT h-chunk partials.
// ---------------------------------------------------------------------------
__global__ void __launch_bounds__(256)
u2_reduce_kernel(const float* __restrict__ part, float* __restrict__ u2) {
  const int k = blockIdx.x * 256 + threadIdx.x;
  float s = 0.0f;
#pragma unroll
  for (int i = 0; i < HSPLIT; ++i) s += part[i * HDIM + k];
  u2[k] = s;
}

// ---------------------------------------------------------------------------
// Phase 2: partial[ks][s] = dot(eo[s, ks*1024 : (ks+1)*1024], u2[...])
// via V_WMMA_F32_16X16X4_F32.  One wave = 16 sequence rows x one K-chunk:
//   A (16x4) : u2[kk..kk+3] broadcast to all 16 M rows
//   B (4x16) : eo rows row0..row0+15, K-slice kk..kk+3
//   D[m][n]  : partial dot for row row0+n (identical for every m)
// 4 independent accumulators break the WMMA C-chain; global_prefetch_b8 pulls
// the eo stream ahead of the b64 load pipeline.
// 2048 waves total (512 blocks x 128 threads) for deep MLP on the 128 MB read.
// ---------------------------------------------------------------------------
__global__ void __launch_bounds__(128)
scores_wmma_kernel(const float* __restrict__ eo, const float* __restrict__ u2,
                   float* __restrict__ partial) {
  const int lane = threadIdx.x & 31;
  const int wave = (blockIdx.x * 128 + (int)threadIdx.x) >> 5;  // [0, 2048)
  const int tile = wave >> 2;        // [0, 512) -> row tile
  const int ks   = wave & 3;         // [0, KSPLIT) -> K chunk
  const int half = lane >> 4;        // 0: K={0,1}, 1: K={2,3}
  const int n    = lane & 15;        // column (sequence row within tile)
  const int row0 = tile * 16;
  const int kb   = ks * KCHUNK;

  const float* __restrict__ rowp = eo + (size_t)(row0 + n) * HDIM + kb + half * 2;
  const float* __restrict__ up   = u2 + kb + half * 2;

  v8f acc0 = {}, acc1 = {}, acc2 = {}, acc3 = {};

  for (int kk = 0; kk < KCHUNK; kk += 16) {
    // Pull the row stream ~1KB ahead (speculative, no LOADcnt cost).
    __builtin_prefetch(rowp + kk + 256, 0, 3);
    v2f a0 = *(const v2f*)(up + kk + 0);
    v2f a1 = *(const v2f*)(up + kk + 4);
    v2f a2 = *(const v2f*)(up + kk + 8);
    v2f a3 = *(const v2f*)(up + kk + 12);
    v2f b0 = *(const v2f*)(rowp + kk + 0);
    v2f b1 = *(const v2f*)(rowp + kk + 4);
    v2f b2 = *(const v2f*)(rowp + kk + 8);
    v2f b3 = *(const v2f*)(rowp + kk + 12);
    acc0 = __builtin_amdgcn_wmma_f32_16x16x4_f32(false, a0, false, b0,
                                                 (short)0, acc0, false, false);
    acc1 = __builtin_amdgcn_wmma_f32_16x16x4_f32(false, a1, false, b1,
                                                 (short)0, acc1, false, false);
    acc2 = __builtin_amdgcn_wmma_f32_16x16x4_f32(false, a2, false, b2,
                                                 (short)0, acc2, false, false);
    acc3 = __builtin_amdgcn_wmma_f32_16x16x4_f32(false, a3, false, b3,
                                                 (short)0, acc3, false, false);
  }

  v8f t = acc0 + acc1 + acc2 + acc3;
  // D VGPR0, lanes 0..15 hold M=0, N=lane -> partial score of row0+lane.
  if (lane < 16) partial[ks * SDIM + row0 + lane] = t[0];
}

// ---------------------------------------------------------------------------
// Phase 3: sum the KSPLIT partials, softmax over 8192 scores.
// One 1024-thread block; scores kept in registers between passes.
// ---------------------------------------------------------------------------
__global__ void __launch_bounds__(1024)
softmax_kernel(const float* __restrict__ partial, float* __restrict__ out) {
  __shared__ float red[1024];
  const int tid = threadIdx.x;
  float v[SDIM / 1024];

  float m = -INFINITY;
#pragma unroll
  for (int j = 0; j < SDIM / 1024; ++j) {
    const int i = tid + j * 1024;
    float s = partial[i] + partial[SDIM + i] + partial[2 * SDIM + i] +
              partial[3 * SDIM + i];
    v[j] = s;
    m = fmaxf(m, s);
  }
  red[tid] = m;
  __syncthreads();
  for (int s = 512; s > 0; s >>= 1) {
    if (tid < s) red[tid] = fmaxf(red[tid], red[tid + s]);
    __syncthreads();
  }
  const float gmax = red[0];
  __syncthreads();

  float sum = 0.0f;
#pragma unroll
  for (int j = 0; j < SDIM / 1024; ++j) {
    float e = __expf(v[j] - gmax);
    v[j] = e;
    sum += e;
  }
  red[tid] = sum;
  __syncthreads();
  for (int s = 512; s > 0; s >>= 1) {
    if (tid < s) red[tid] += red[tid + s];
    __syncthreads();
  }
  const float inv = 1.0f / red[0];
#pragma unroll
  for (int j = 0; j < SDIM / 1024; ++j) out[tid + j * 1024] = v[j] * inv;
}

// ---------------------------------------------------------------------------
// Launcher.  Softmax shift-invariance removes the hidden/bias contribution
// (constant across s), so d_in[0] (hidden) and d_in[3] (attn_b) are unused.
// ---------------------------------------------------------------------------
extern "C" void kernel_launch(void* const* d_in, const int* in_sizes, int n_in,
                              void* d_out, int out_size, void* d_ws, size_t ws_size,
                              hipStream_t stream) {
  (void)in_sizes; (void)n_in; (void)out_size; (void)ws_size;
  const float* hidden = (const float*)d_in[0];  (void)hidden;   // drops out of softmax
  const float* eo     = (const float*)d_in[1];                  // (S, 1, H) contiguous
  const float* W      = (const float*)d_in[2];                  // (H, 2H) row-major
  const float* bias   = (const float*)d_in[3];  (void)bias;     // drops out of softmax
  const float* other  = (const float*)d_in[4];

  float* ws      = (float*)d_ws;
  float* part    = ws;                       // HSPLIT * 4096 floats (256 KB)
  float* u2      = part + HSPLIT * HDIM;     // 4096 floats
  float* partial = u2 + HDIM;                // KSPLIT * 8192 floats (128 KB)
  float* out     = (float*)d_out;            // 8192 floats (1,1,S)

  dim3 g1(16, HSPLIT);
  u2_partial_kernel<<<g1, 256, 0, stream>>>(W, other, part);
  u2_reduce_kernel<<<16, 256, 0, stream>>>(part, u2);
  scores_wmma_kernel<<<512, 128, 0, stream>>>(eo, u2, partial);
  softmax_kernel<<<1, 1024, 0, stream>>>(partial, out);
}